// LeapfrogIntegrator_2370821948212
// MI455X (gfx1250) — compile-verified
//
#include <hip/hip_runtime.h>

typedef __attribute__((ext_vector_type(2))) float v2f;
typedef __attribute__((ext_vector_type(8))) float v8f;
typedef __attribute__((ext_vector_type(4))) unsigned int tdm_g0_t;
typedef __attribute__((ext_vector_type(8))) int tdm_v8i_t;
typedef __attribute__((ext_vector_type(4))) int tdm_v4i_t;

#define WMMA_F32(a, b, c) \
  __builtin_amdgcn_wmma_f32_16x16x4_f32(false, (a), false, (b), (short)0, (c), false, false)

constexpr int   kB  = 8192;
constexpr int   kD  = 1024;
constexpr int   kR  = 256;
constexpr float kDt = 0.01f;          // DT * DT_SCALE
constexpr int   KC  = 32;             // K-chunk per TDM transfer
constexpr int   LP  = KC + 2;         // LDS row pitch (dwords): TDM pads +2/row -> conflict-free b64

// ----------------------------------------------------------------------------
// TDM: DMA a 64-row x 32-col fp32 tile (global row stride = row_elems) into
// LDS at lds_byte, +2 dwords pad per 32-dword row (pitch 34 dwords).
// D# layout per CDNA5 ISA 8.3-8.6 (2D tensor, groups 2/3 zero).
// ----------------------------------------------------------------------------
__device__ __forceinline__
void tdm_load_tile64x32(const float* g, unsigned lds_byte, unsigned row_elems) {
  unsigned long long ga = (unsigned long long)(uintptr_t)g;
  tdm_g0_t g0;
  g0[0] = 1u;                                   // count=1 (valid user load)
  g0[1] = lds_byte;                             // lds_addr (bytes)
  g0[2] = (unsigned)ga;                         // global_addr[31:0]
  g0[3] = (unsigned)(ga >> 32) | (2u << 30);    // global_addr[56:32] | type=2
  const unsigned td0 = row_elems;               // tensor_dim0 (tile is interior)
  const unsigned td1 = 1u << 20;                // tensor_dim1 (tile is interior)
  tdm_v8i_t g1;
  g1[0] = (int)((2u << 16)                      // data_size = 4 bytes
              | (1u << 20)                      // pad_enable
              | (4u << 22)                      // pad_interval: every 32 dwords
              | (1u << 25));                    // pad_amount: 2 dwords
  g1[1] = (int)((td0 & 0xFFFFu) << 16);
  g1[2] = (int)((td0 >> 16) | ((td1 & 0xFFFFu) << 16));
  g1[3] = (int)((td1 >> 16) | ((unsigned)KC << 16));   // tile_dim0 = 32
  g1[4] = 64;                                          // tile_dim1 = 64, tile_dim2 = 0
  g1[5] = (int)row_elems;                              // tensor_dim0_stride lo32
  g1[6] = 0;
  g1[7] = 0;
  tdm_v4i_t z4 = {0, 0, 0, 0};
  tdm_v8i_t z8 = {0, 0, 0, 0, 0, 0, 0, 0};
  __builtin_amdgcn_tensor_load_to_lds(g0, g1, z4, z4, z8, 0);
}

// ----------------------------------------------------------------------------
// s[b,r] = ( sum_d vin[b,d] * U[r,d] )^2
// Wave tile 16(M=b) x 64(N=r); 8 waves stacked in M -> block 128 x 64.
// B tile (U rows n0..n0+63) TDM-staged in LDS, double buffered, shared by all
// 8 waves.  grid = (kR/64, kB/128)
// ----------------------------------------------------------------------------
__global__ __launch_bounds__(256)
void k_uu(const float* __restrict__ vin, const float* __restrict__ U,
          float* __restrict__ s) {
  __shared__ __align__(16) float sm[2][64 * LP];
  const int lane = threadIdx.x & 31;
  const int wave = threadIdx.x >> 5;
  const int m0   = blockIdx.y * 128 + wave * 16;
  const int n0   = blockIdx.x * 64;
  const int mr   = lane & 15;
  const int koff = (lane >> 4) << 1;

  const float* a = vin + (size_t)(m0 + mr) * kD + koff;

  v8f c0 = {}, c1 = {}, c2 = {}, c3 = {};

  if (wave == 0)
    tdm_load_tile64x32(U + (size_t)n0 * kD, (unsigned)(uintptr_t)&sm[0][0], kD);
  __builtin_amdgcn_s_wait_tensorcnt(0);
  __syncthreads();

  constexpr int NCH = kD / KC;
  for (int c = 0; c < NCH; ++c) {
    if (wave == 0 && c + 1 < NCH)
      tdm_load_tile64x32(U + (size_t)n0 * kD + (size_t)(c + 1) * KC,
                         (unsigned)(uintptr_t)&sm[(c + 1) & 1][0], kD);
    const float* sb = &sm[c & 1][0];
    const int kb = c * KC;
#pragma unroll
    for (int k = 0; k < KC; k += 4) {
      v2f af = *(const v2f*)(a + kb + k);
      v2f b0 = *(const v2f*)(sb + ( 0 + mr) * LP + k + koff);
      v2f b1 = *(const v2f*)(sb + (16 + mr) * LP + k + koff);
      v2f b2 = *(const v2f*)(sb + (32 + mr) * LP + k + koff);
      v2f b3 = *(const v2f*)(sb + (48 + mr) * LP + k + koff);
      c0 = WMMA_F32(af, b0, c0);
      c1 = WMMA_F32(af, b1, c1);
      c2 = WMMA_F32(af, b2, c2);
      c3 = WMMA_F32(af, b3, c3);
    }
    __builtin_amdgcn_s_wait_tensorcnt(0);   // no-op for waves 1..7
    __syncthreads();
  }

  const int rbase = m0 + ((lane >> 4) << 3);
#pragma unroll
  for (int i = 0; i < 8; ++i) {
    size_t o = (size_t)(rbase + i) * kR + n0 + mr;
    s[o +  0] = c0[i] * c0[i];
    s[o + 16] = c1[i] * c1[i];
    s[o + 32] = c2[i] * c2[i];
    s[o + 48] = c3[i] * c3[i];
  }
}

// ----------------------------------------------------------------------------
// gamma[b,d] = sum_r s[b,r] * W[d,r]   (W : [kD, kR] row-major), fused update:
//   phase 1:  vh = v + 0.5*dt*(force - gamma); out_x = x + dt*vh; out_v = vh
//   phase 2:  out_v = vbase + 0.5*dt*(force - gamma)      (vbase == out_v, ok)
// grid = (kD/64, kB/128)
// ----------------------------------------------------------------------------
__global__ __launch_bounds__(256)
void k_gamma(const float* __restrict__ s, const float* __restrict__ W,
             const float* __restrict__ vbase, const float* __restrict__ force,
             const float* __restrict__ x, float* __restrict__ out_x,
             float* __restrict__ out_v, int phase) {
  __shared__ __align__(16) float sm[2][64 * LP];
  const int lane = threadIdx.x & 31;
  const int wave = threadIdx.x >> 5;
  const int m0   = blockIdx.y * 128 + wave * 16;
  const int n0   = blockIdx.x * 64;
  const int mr   = lane & 15;
  const int koff = (lane >> 4) << 1;

  const float* a = s + (size_t)(m0 + mr) * kR + koff;

  v8f c0 = {}, c1 = {}, c2 = {}, c3 = {};

  if (wave == 0)
    tdm_load_tile64x32(W + (size_t)n0 * kR, (unsigned)(uintptr_t)&sm[0][0], kR);
  __builtin_amdgcn_s_wait_tensorcnt(0);
  __syncthreads();

  constexpr int NCH = kR / KC;
  for (int c = 0; c < NCH; ++c) {
    if (wave == 0 && c + 1 < NCH)
      tdm_load_tile64x32(W + (size_t)n0 * kR + (size_t)(c + 1) * KC,
                         (unsigned)(uintptr_t)&sm[(c + 1) & 1][0], kR);
    const float* sb = &sm[c & 1][0];
    const int kb = c * KC;
#pragma unroll
    for (int k = 0; k < KC; k += 4) {
      v2f af = *(const v2f*)(a + kb + k);
      v2f b0 = *(const v2f*)(sb + ( 0 + mr) * LP + k + koff);
      v2f b1 = *(const v2f*)(sb + (16 + mr) * LP + k + koff);
      v2f b2 = *(const v2f*)(sb + (32 + mr) * LP + k + koff);
      v2f b3 = *(const v2f*)(sb + (48 + mr) * LP + k + koff);
      c0 = WMMA_F32(af, b0, c0);
      c1 = WMMA_F32(af, b1, c1);
      c2 = WMMA_F32(af, b2, c2);
      c3 = WMMA_F32(af, b3, c3);
    }
    __builtin_amdgcn_s_wait_tensorcnt(0);
    __syncthreads();
  }

  const float halfdt = 0.5f * kDt;
  const int rbase = m0 + ((lane >> 4) << 3);
#pragma unroll
  for (int i = 0; i < 8; ++i) {
    size_t base = (size_t)(rbase + i) * kD + n0 + mr;
    float g[4] = {c0[i], c1[i], c2[i], c3[i]};
#pragma unroll
    for (int nn = 0; nn < 4; ++nn) {
      size_t idx = base + (size_t)(16 * nn);
      float vh = vbase[idx] + halfdt * (force[idx] - g[nn]);
      if (phase == 1) {
        out_x[idx] = x[idx] + kDt * vh;
      }
      out_v[idx] = vh;
    }
  }
}

extern "C" void kernel_launch(void* const* d_in, const int* in_sizes, int n_in,
                              void* d_out, int out_size, void* d_ws, size_t ws_size,
                              hipStream_t stream) {
  (void)in_sizes; (void)n_in; (void)out_size; (void)ws_size;
  const float* x     = (const float*)d_in[0];  // [B, D]
  const float* v     = (const float*)d_in[1];  // [B, D]
  const float* force = (const float*)d_in[2];  // [B, D]
  const float* U     = (const float*)d_in[3];  // [R, D]
  const float* W     = (const float*)d_in[4];  // [D, R]

  float* x_new = (float*)d_out;                    // [B, D]
  float* v_out = (float*)d_out + (size_t)kB * kD;  // v_half then v_new
  float* sws   = (float*)d_ws;                     // [B, R] scratch (8 MB)

  dim3 blk(256, 1, 1);
  dim3 g1(kR / 64, kB / 128, 1);   // (4, 64)
  dim3 g2(kD / 64, kB / 128, 1);   // (16, 64)

  k_uu   <<<g1, blk, 0, stream>>>(v, U, sws);
  k_gamma<<<g2, blk, 0, stream>>>(sws, W, v, force, x, x_new, v_out, 1);
  k_uu   <<<g1, blk, 0, stream>>>(v_out, U, sws);
  k_gamma<<<g2, blk, 0, stream>>>(sws, W, v_out, force, x, x_new, v_out, 2);
}